// GRU_GCN_76055280877942
// MI455X (gfx1250) — compile-verified
//
#include <hip/hip_runtime.h>

#define TPB 256
#define EPSV 1e-5f
#define NEG_SLOPE 0.01f

typedef __attribute__((ext_vector_type(2))) float v2f;
typedef __attribute__((ext_vector_type(8))) float v8f;

__device__ __forceinline__ float sigmoidf_(float x) { return 1.0f / (1.0f + __expf(-x)); }
__device__ __forceinline__ float leaky_(float x) { return x >= 0.0f ? x : NEG_SLOPE * x; }

// ---------------- init: deg = 1.0 (self loop weight) ----------------
__global__ void k_init_deg(float* __restrict__ deg, int N) {
    int i = blockIdx.x * blockDim.x + threadIdx.x;
    if (i < N) deg[i] = 1.0f;
}

// ---------------- degree scatter over edges ----------------
__global__ void k_deg(const int* __restrict__ ei, const float* __restrict__ w, int E,
                      float* __restrict__ deg) {
    int e = blockIdx.x * blockDim.x + threadIdx.x;
    if (e >= E) return;
    atomicAdd(&deg[ei[E + e]], w[e]);   // col = ei[E..2E)
}

__global__ void k_dinv(const float* __restrict__ deg, float* __restrict__ dinv, int N) {
    int i = blockIdx.x * blockDim.x + threadIdx.x;
    if (i >= N) return;
    float d = deg[i];
    dinv[i] = d > 0.0f ? rsqrtf(fmaxf(d, EPSV)) : 0.0f;
}

// ---------------- per-node GRU (seq=4, in=1, hid=2) -> x1_pre[N,9] ----------------
__global__ void k_gru(const float* __restrict__ x,
                      const float* __restrict__ wih, const float* __restrict__ whh,
                      const float* __restrict__ bih, const float* __restrict__ bhh,
                      int N, float* __restrict__ x1p) {
    int i = blockIdx.x * blockDim.x + threadIdx.x;
    if (i >= N) return;
    float h0 = 0.0f, h1 = 0.0f;
    float o[9];
#pragma unroll
    for (int t = 0; t < 4; ++t) {
        float xt = x[i * 5 + t];
        float r0 = sigmoidf_(xt * wih[0] + bih[0] + h0 * whh[0] + h1 * whh[1] + bhh[0]);
        float r1 = sigmoidf_(xt * wih[1] + bih[1] + h0 * whh[2] + h1 * whh[3] + bhh[1]);
        float z0 = sigmoidf_(xt * wih[2] + bih[2] + h0 * whh[4] + h1 * whh[5] + bhh[2]);
        float z1 = sigmoidf_(xt * wih[3] + bih[3] + h0 * whh[6] + h1 * whh[7] + bhh[3]);
        float gi4 = xt * wih[4] + bih[4], gi5 = xt * wih[5] + bih[5];
        float gh4 = h0 * whh[8] + h1 * whh[9] + bhh[4];
        float gh5 = h0 * whh[10] + h1 * whh[11] + bhh[5];
        float n0 = tanhf(gi4 + r0 * gh4);
        float n1 = tanhf(gi5 + r1 * gh5);
        h0 = (1.0f - z0) * n0 + z0 * h0;
        h1 = (1.0f - z1) * n1 + z1 * h1;
        o[2 * t] = h0; o[2 * t + 1] = h1;
    }
    o[8] = x[i * 5 + 4];
#pragma unroll
    for (int c = 0; c < 9; ++c) x1p[(size_t)i * 9 + c] = o[c];
}

// ---------------- per-channel row statistics (deterministic tree reduce) --------
// MODE 0: sum of (src+bias)   MODE 1: sum of ((src+bias) - ms*mean)^2
template <int C, int MODE>
__global__ void k_rowstat(const float* __restrict__ src, const float* __restrict__ bias,
                          const float* __restrict__ mean, const float* __restrict__ ms,
                          int N, float* __restrict__ partials) {
    __shared__ float sd[TPB * C];
    int tid = threadIdx.x;
    int i = blockIdx.x * TPB + tid;
    float v[C];
    if (i < N) {
#pragma unroll
        for (int c = 0; c < C; ++c) {
            float t = src[(size_t)i * C + c] + (bias ? bias[c] : 0.0f);
            if (MODE) { t = t - ms[c] * mean[c]; t = t * t; }
            v[c] = t;
        }
    } else {
#pragma unroll
        for (int c = 0; c < C; ++c) v[c] = 0.0f;
    }
#pragma unroll
    for (int c = 0; c < C; ++c) sd[tid * C + c] = v[c];
    __syncthreads();
    for (int s = TPB / 2; s > 0; s >>= 1) {
        if (tid < s)
#pragma unroll
            for (int c = 0; c < C; ++c) sd[tid * C + c] += sd[(tid + s) * C + c];
        __syncthreads();
    }
    if (tid < C) partials[blockIdx.x * 16 + tid] = sd[tid];
}

// mode 0: out[c] = sum/N     mode 1: out[c] = rsqrt(sum/N + eps)
__global__ void k_reduce(const float* __restrict__ partials, int nblocks, int C,
                         float invN, int mode, float* __restrict__ out) {
    int c = threadIdx.x;
    if (c >= C) return;
    float s = 0.0f;
    for (int b = 0; b < nblocks; ++b) s += partials[b * 16 + c];
    float m = s * invN;
    out[c] = mode ? rsqrtf(m + EPSV) : m;
}

// ------- fused GraphNorm1-apply + LeakyReLU + (x1 @ gcn_w) via V_WMMA_F32_16X16X4_F32
// One wave handles a 16-node tile; K=9 padded to 12 -> 3 chained WMMAs.
// Also writes the self-loop message dinv^2 * h directly into x2 (full coverage).
__global__ void k_x1h_wmma(const float* __restrict__ x1p,
                           const float* __restrict__ gn1w, const float* __restrict__ gn1b,
                           const float* __restrict__ gn1ms,
                           const float* __restrict__ stats, // mean1 @ 0, rstd1 @ 16
                           const float* __restrict__ gcnw, const float* __restrict__ dinv,
                           int N, int ntiles,
                           float* __restrict__ h, float* __restrict__ x2) {
    int gt = blockIdx.x * blockDim.x + threadIdx.x;
    int wave = gt >> 5;
    int lane = threadIdx.x & 31;
    if (wave >= ntiles) return;          // wave-uniform: EXEC stays all-1s for WMMA
    int hh = lane >> 4;                  // lane half (selects K pair / output row block)
    int m = lane & 15;                   // A-row within tile / D-column index
    int base = wave * 16;
    int node = base + m;
    if (node >= N) node = N - 1;         // clamped read (writes are guarded)
    const float* row = x1p + (size_t)node * 9;

    auto x1n = [&](int k) -> float {     // GraphNorm1 + LeakyReLU, zero-padded K
        if (k >= 9) return 0.0f;
        float c = row[k] - gn1ms[k] * stats[k];
        float y = gn1w[k] * c * stats[16 + k] + gn1b[k];
        return leaky_(y);
    };

    int n = m;                           // B/D column (valid cols 0..7)
    v8f acc = {};
#pragma unroll
    for (int kk = 0; kk < 12; kk += 4) {
        int k0 = kk + 2 * hh;
        v2f a; a.x = x1n(k0); a.y = x1n(k0 + 1);
        v2f b;
        b.x = (k0 < 9 && n < 8) ? gcnw[k0 * 8 + n] : 0.0f;
        b.y = (k0 + 1 < 9 && n < 8) ? gcnw[(k0 + 1) * 8 + n] : 0.0f;
        acc = __builtin_amdgcn_wmma_f32_16x16x4_f32(false, a, false, b,
                                                    (short)0, acc, false, false);
    }
    if (n < 8) {
#pragma unroll
        for (int v = 0; v < 8; ++v) {    // D: VGPR v holds rows v (lanes 0-15) / v+8 (16-31)
            int nd = base + v + 8 * hh;
            if (nd < N) {
                float hv = acc[v];
                h[(size_t)nd * 8 + n] = hv;
                float di = dinv[nd];
                x2[(size_t)nd * 8 + n] = di * di * hv;  // self-loop message
            }
        }
    }
}

// ---------------- edge message scatter (L2-resident h/x2) ----------------
__global__ void k_edges(const int* __restrict__ ei, const float* __restrict__ w,
                        const float* __restrict__ dinv, const float* __restrict__ h,
                        int E, float* __restrict__ x2) {
    int e = blockIdx.x * blockDim.x + threadIdx.x;
    if (e >= E) return;
    int r = ei[e];
    int c = ei[E + e];
    float nrm = dinv[r] * w[e] * dinv[c];
    const float4* hr = (const float4*)(h + (size_t)r * 8);
    float4 a = hr[0], b = hr[1];
    float* dst = x2 + (size_t)c * 8;
    atomicAdd(dst + 0, nrm * a.x); atomicAdd(dst + 1, nrm * a.y);
    atomicAdd(dst + 2, nrm * a.z); atomicAdd(dst + 3, nrm * a.w);
    atomicAdd(dst + 4, nrm * b.x); atomicAdd(dst + 5, nrm * b.y);
    atomicAdd(dst + 6, nrm * b.z); atomicAdd(dst + 7, nrm * b.w);
}

// ---------------- GraphNorm2-apply + LeakyReLU + FC(8->1) ----------------
__global__ void k_final(const float* __restrict__ x2, const float* __restrict__ gcnb,
                        const float* __restrict__ gn2w, const float* __restrict__ gn2b,
                        const float* __restrict__ gn2ms,
                        const float* __restrict__ stats, // mean2 @ 32, rstd2 @ 48
                        const float* __restrict__ fcw, const float* __restrict__ fcb,
                        int N, float* __restrict__ out) {
    int i = blockIdx.x * blockDim.x + threadIdx.x;
    if (i >= N) return;
    float s = 0.0f;
#pragma unroll
    for (int c = 0; c < 8; ++c) {
        float t = x2[(size_t)i * 8 + c] + gcnb[c];
        t = t - gn2ms[c] * stats[32 + c];
        float y = gn2w[c] * t * stats[48 + c] + gn2b[c];
        s += leaky_(y) * fcw[c];
    }
    out[i] = s + fcb[0];
}

extern "C" void kernel_launch(void* const* d_in, const int* in_sizes, int n_in,
                              void* d_out, int out_size, void* d_ws, size_t ws_size,
                              hipStream_t stream) {
    const float* x     = (const float*)d_in[0];
    const int*   ei    = (const int*)  d_in[1];
    const float* ew    = (const float*)d_in[2];
    const float* wih   = (const float*)d_in[3];
    const float* whh   = (const float*)d_in[4];
    const float* bih   = (const float*)d_in[5];
    const float* bhh   = (const float*)d_in[6];
    const float* gn1w  = (const float*)d_in[7];
    const float* gn1b  = (const float*)d_in[8];
    const float* gn1ms = (const float*)d_in[9];
    const float* gn2w  = (const float*)d_in[10];
    const float* gn2b  = (const float*)d_in[11];
    const float* gn2ms = (const float*)d_in[12];
    const float* gcnw  = (const float*)d_in[13];
    const float* gcnb  = (const float*)d_in[14];
    const float* fcw   = (const float*)d_in[15];
    const float* fcb   = (const float*)d_in[16];

    const int N = in_sizes[0] / 5;
    const int E = in_sizes[2];

    // workspace layout (floats, 16-elem aligned regions)
    float* ws = (float*)d_ws;
    size_t off = 0;
    auto take = [&](size_t n) { size_t r = off; off = (off + n + 15) & ~(size_t)15; return ws + r; };
    float* x1p   = take((size_t)N * 9);
    float* hbuf  = take((size_t)N * 8);
    float* x2    = take((size_t)N * 8);
    float* deg   = take((size_t)N);
    float* dinv  = take((size_t)N);
    const int nbN = (N + TPB - 1) / TPB;
    const int nbE = (E + TPB - 1) / TPB;
    float* part  = take((size_t)nbN * 16);
    float* stats = take(64); // mean1[0..8], rstd1[16..24], mean2[32..39], rstd2[48..55]

    const float invN = 1.0f / (float)N;

    // degree / dinv path
    k_init_deg<<<nbN, TPB, 0, stream>>>(deg, N);
    k_deg<<<nbE, TPB, 0, stream>>>(ei, ew, E, deg);
    k_dinv<<<nbN, TPB, 0, stream>>>(deg, dinv, N);

    // GRU -> x1_pre ; GraphNorm1 stats
    k_gru<<<nbN, TPB, 0, stream>>>(x, wih, whh, bih, bhh, N, x1p);
    k_rowstat<9, 0><<<nbN, TPB, 0, stream>>>(x1p, nullptr, nullptr, nullptr, N, part);
    k_reduce<<<1, 32, 0, stream>>>(part, nbN, 9, invN, 0, stats + 0);
    k_rowstat<9, 1><<<nbN, TPB, 0, stream>>>(x1p, nullptr, stats + 0, gn1ms, N, part);
    k_reduce<<<1, 32, 0, stream>>>(part, nbN, 9, invN, 1, stats + 16);

    // fused norm1 + leaky + GEMM (WMMA f32 16x16x4) + self-loop into x2
    const int ntiles = (N + 15) / 16;
    const int nbW = (ntiles * 32 + TPB - 1) / TPB;
    k_x1h_wmma<<<nbW, TPB, 0, stream>>>(x1p, gn1w, gn1b, gn1ms, stats, gcnw, dinv,
                                        N, ntiles, hbuf, x2);

    // edge scatter
    k_edges<<<nbE, TPB, 0, stream>>>(ei, ew, dinv, hbuf, E, x2);

    // GraphNorm2 stats (over x2 + gcn_b)
    k_rowstat<8, 0><<<nbN, TPB, 0, stream>>>(x2, gcnb, nullptr, nullptr, N, part);
    k_reduce<<<1, 32, 0, stream>>>(part, nbN, 8, invN, 0, stats + 32);
    k_rowstat<8, 1><<<nbN, TPB, 0, stream>>>(x2, gcnb, stats + 32, gn2ms, N, part);
    k_reduce<<<1, 32, 0, stream>>>(part, nbN, 8, invN, 1, stats + 48);

    // norm2 + leaky + FC
    k_final<<<nbN, TPB, 0, stream>>>(x2, gcnb, gn2w, gn2b, gn2ms, stats, fcw, fcb,
                                     N, (float*)d_out);
}